// RandomizedTopKBaseline_30030411334098
// MI455X (gfx1250) — compile-verified
//
#include <hip/hip_runtime.h>
#include <math.h>

#ifndef __has_builtin
#define __has_builtin(x) 0
#endif

// ---------------- problem constants ----------------
#define B_DIM  32
#define N_ROW  262144                 // 256*32*32
#define N_TOT  (B_DIM * N_ROW)        // 8388608
#define K_SEL  13107u                 // int(0.05 * N_ROW)
#define BETA_DIV 0.100001f            // EPS + 1e-6

#define SPLIT_A 16
#define CHUNK_A (N_ROW / SPLIT_A)     // 16384
#define SPLIT_H 32
#define CHUNK_H (N_ROW / SPLIT_H)     // 8192
#define ITERS_H (CHUNK_H / 1024)      // 8  (256 thr * float4)

// ---------------- workspace layout (bytes) ----------------
#define WS_PART 0                     // 512*2 floats  (block partial sums)
#define WS_BETA 4096                  // 32 floats
#define WS_RANK 4224                  // 32 u32
#define WS_PREF 4352                  // 32 u32
#define WS_THR  4480                  // 32 u32
#define WS_H1   4608                  // 32*4096 u32 = 512 KB
#define WS_H2   (4608 + 524288)
#define WS_H3   (4608 + 1048576)      // 32*256 u32 = 32 KB
#define WS_USED (4608 + 1048576 + 32768)

typedef __attribute__((ext_vector_type(2))) float v2f;
typedef __attribute__((ext_vector_type(8))) float v8f;

// ---------------- CDNA5 async LDS staging (inline asm, toolchain-portable) ---
__device__ __forceinline__ void async_load_b128(void* lds, const void* gaddr) {
    // low 32 bits of a generic shared pointer == LDS byte offset (aperture encoding)
    unsigned l = (unsigned)(size_t)lds;
    asm volatile("global_load_async_to_lds_b128 %0, %1, off"
                 :: "v"(l), "v"(gaddr) : "memory");
}
__device__ __forceinline__ void wait_async_le2() {
    asm volatile("s_wait_asynccnt 2" ::: "memory");
}
__device__ __forceinline__ void wait_async_0() {
    asm volatile("s_wait_asynccnt 0" ::: "memory");
}

// ---------------- WMMA-based block reduction (256 -> 1, deterministic) ------
// D = A x Ones + C : row-sums of A; 4 accumulating WMMAs eat 256 partials.
__device__ __forceinline__ float block_reduce_sum(float v, float* red) {
    const int tid = threadIdx.x;
    red[tid] = v;
    __syncthreads();
    float total = 0.0f;
    if (tid < 32) {   // wave 0, EXEC all-ones for this wave
#if __has_builtin(__builtin_amdgcn_wmma_f32_16x16x4_f32)
        v8f c = {};
        v2f bones; bones[0] = 1.0f; bones[1] = 1.0f;
#pragma unroll
        for (int t = 0; t < 4; ++t) {
            v2f a; a[0] = red[t * 64 + tid]; a[1] = red[t * 64 + 32 + tid];
            c = __builtin_amdgcn_wmma_f32_16x16x4_f32(
                    false, a, false, bones, (short)0, c, false, false);
        }
        float s = 0.0f;
#pragma unroll
        for (int i = 0; i < 8; ++i) s += c[i];
        s += __shfl_xor(s, 16, 32);   // lanes 0-15 hold rows 0-7, 16-31 rows 8-15
        total = s;
#else
        float s = 0.0f;
#pragma unroll
        for (int j = 0; j < 8; ++j) s += red[tid + 32 * j];
        s += __shfl_xor(s, 16, 32);
        s += __shfl_xor(s, 8, 32);
        s += __shfl_xor(s, 4, 32);
        s += __shfl_xor(s, 2, 32);
        s += __shfl_xor(s, 1, 32);
        total = s;
#endif
    }
    __syncthreads();
    return total;
}

// ---------------- pass 1a: per-block sum / sumsq -----------------------------
__global__ void __launch_bounds__(256)
kSum1(const float* __restrict__ x, float* __restrict__ part) {
    const int tid = threadIdx.x;
    const int row = blockIdx.x / SPLIT_A, p = blockIdx.x % SPLIT_A;
    const float4* xv = (const float4*)(x + (size_t)row * N_ROW + (size_t)p * CHUNK_A);
    float s = 0.0f, ss = 0.0f;
#pragma unroll 4
    for (int i = 0; i < CHUNK_A / 1024; ++i) {          // 16 float4 per thread
        float4 v = xv[i * 256 + tid];
        s  += v.x + v.y + v.z + v.w;
        ss += v.x * v.x + v.y * v.y + v.z * v.z + v.w * v.w;
    }
    __shared__ float red[256];
    float ts  = block_reduce_sum(s,  red);
    float tss = block_reduce_sum(ss, red);
    if (tid == 0) {
        part[blockIdx.x * 2 + 0] = ts;
        part[blockIdx.x * 2 + 1] = tss;
    }
}

// ---------------- pass 1b: beta per row (fixed-order, deterministic) ---------
__global__ void kSum2(const float* __restrict__ part, float* __restrict__ beta) {
    const int b = threadIdx.x;      // 32 threads
    float s = 0.0f, ss = 0.0f;
    for (int p = 0; p < SPLIT_A; ++p) {
        s  += part[(b * SPLIT_A + p) * 2 + 0];
        ss += part[(b * SPLIT_A + p) * 2 + 1];
    }
    const float n = (float)N_ROW;
    float var = (ss - s * s / n) / (n - 1.0f);
    beta[b] = sqrtf(fmaxf(var, 0.0f)) / BETA_DIV;
}

// ---------------- pass 2: acts + level-1 histogram (async LDS pipeline) -----
__global__ void __launch_bounds__(256)
kActsHist1(const float* __restrict__ x, const float* __restrict__ u,
           const float* __restrict__ beta, float* __restrict__ acts,
           unsigned* __restrict__ hist) {
    __shared__ float sx[2][1024];
    __shared__ float su[2][1024];
    __shared__ unsigned lh[4096];
    const int tid = threadIdx.x;
    const int row = blockIdx.x / SPLIT_H, p = blockIdx.x % SPLIT_H;
    const size_t base = (size_t)row * N_ROW + (size_t)p * CHUNK_H;

    for (int i = tid; i < 4096; i += 256) lh[i] = 0;
    __syncthreads();
    const float bet = beta[row];

    // prologue: async prefetch tile 0 into buffer 0
    async_load_b128(&sx[0][tid * 4], x + base + tid * 4);
    async_load_b128(&su[0][tid * 4], u + base + tid * 4);

    for (int it = 0; it < ITERS_H; ++it) {
        if (it + 1 < ITERS_H) {
            const int nb = (it + 1) & 1;
            const size_t off = base + (size_t)(it + 1) * 1024 + tid * 4;
            async_load_b128(&sx[nb][tid * 4], x + off);
            async_load_b128(&su[nb][tid * 4], u + off);
            wait_async_le2();          // 2 newest may stay in flight => tile 'it' done
        } else {
            wait_async_0();
        }
        const int cb = it & 1;
        float4 xv = *(const float4*)&sx[cb][tid * 4];
        float4 uv = *(const float4*)&su[cb][tid * 4];

        float a[4];
        {
            float xc[4] = {xv.x, xv.y, xv.z, xv.w};
            float uc[4] = {uv.x, uv.y, uv.z, uv.w};
#pragma unroll
            for (int j = 0; j < 4; ++j) {
                float g = -__logf(-__logf(uc[j]));
                a[j] = fmaxf(fmaf(bet, g, xc[j]), 0.0f);
            }
        }
        float4 av = make_float4(a[0], a[1], a[2], a[3]);
        *(float4*)(acts + base + (size_t)it * 1024 + tid * 4) = av;
#pragma unroll
        for (int j = 0; j < 4; ++j)
            atomicAdd(&lh[__float_as_uint(a[j]) >> 20], 1u);
    }
    __syncthreads();
    unsigned* gh = hist + (size_t)row * 4096;
    for (int i = tid; i < 4096; i += 256) {
        unsigned c = lh[i];
        if (c) atomicAdd(&gh[i], c);
    }
}

// ---------------- level-2/3 histograms (read cached acts) --------------------
__global__ void __launch_bounds__(256)
kHistL(const float* __restrict__ acts, const unsigned* __restrict__ pref,
       unsigned* __restrict__ hist, int bins, int cmpShift, int binShift,
       unsigned binMask) {
    __shared__ unsigned lh[4096];
    const int tid = threadIdx.x;
    const int row = blockIdx.x / SPLIT_H, p = blockIdx.x % SPLIT_H;
    const size_t base = (size_t)row * N_ROW + (size_t)p * CHUNK_H;
    for (int i = tid; i < bins; i += 256) lh[i] = 0;
    __syncthreads();
    const unsigned pr = pref[row];
    for (int it = 0; it < ITERS_H; ++it) {
        float4 av = *(const float4*)(acts + base + (size_t)it * 1024 + tid * 4);
        float a[4] = {av.x, av.y, av.z, av.w};
#pragma unroll
        for (int j = 0; j < 4; ++j) {
            unsigned bits = __float_as_uint(a[j]);
            if ((bits >> cmpShift) == pr)
                atomicAdd(&lh[(bits >> binShift) & binMask], 1u);
        }
    }
    __syncthreads();
    unsigned* gh = hist + (size_t)row * bins;
    for (int i = tid; i < bins; i += 256) {
        unsigned c = lh[i];
        if (c) atomicAdd(&gh[i], c);
    }
}

// ---------------- per-row top-down scan of a histogram level -----------------
__global__ void __launch_bounds__(256)
kScan(const unsigned* __restrict__ hist, unsigned* __restrict__ rank,
      unsigned* __restrict__ pref, unsigned* __restrict__ thr,
      int bins, int level) {
    __shared__ unsigned sh[4096];
    __shared__ unsigned csums[256];
    __shared__ unsigned suff[257];
    const int row = blockIdx.x, tid = threadIdx.x;
    const unsigned* h = hist + (size_t)row * bins;
    const int chunk = bins / 256;                 // 16 or 1
    for (int i = tid; i < bins; i += 256) sh[i] = h[i];
    __syncthreads();
    unsigned cs = 0;
    for (int j = 0; j < chunk; ++j) cs += sh[tid * chunk + j];
    csums[tid] = cs;
    __syncthreads();
    if (tid == 0) {
        unsigned acc = 0; suff[256] = 0;
        for (int i = 255; i >= 0; --i) { acc += csums[i]; suff[i] = acc; }
    }
    __syncthreads();
    const unsigned kk = (level == 1) ? K_SEL : rank[row];
    // exactly one thread owns the crossing chunk
    if (suff[tid] >= kk && suff[tid + 1] < kk) {
        unsigned cum = suff[tid + 1];
        int bsel = tid * chunk;
        unsigned cnt = 0;
        for (int j = chunk - 1; j >= 0; --j) {
            cnt = sh[tid * chunk + j];
            cum += cnt;
            if (cum >= kk) { bsel = tid * chunk + j; break; }
        }
        rank[row] = kk - (cum - cnt);             // rank within selected bin
        if (level == 1)      pref[row] = (unsigned)bsel;
        else if (level == 2) pref[row] = (pref[row] << 12) | (unsigned)bsel;
        else                 thr[row]  = (pref[row] << 8)  | (unsigned)bsel;
    }
}

// ---------------- final: mask + sparse (in-place over acts) ------------------
__global__ void __launch_bounds__(256)
kFinal(float* __restrict__ acts, const unsigned* __restrict__ thr,
       float* __restrict__ mask) {
    const int tid = threadIdx.x;
    const int row = blockIdx.x / SPLIT_H, p = blockIdx.x % SPLIT_H;
    const size_t base = (size_t)row * N_ROW + (size_t)p * CHUNK_H;
    const unsigned T = thr[row];
    for (int it = 0; it < ITERS_H; ++it) {
        const size_t idx = base + (size_t)it * 1024 + tid * 4;
        float4 av = *(const float4*)(acts + idx);
        float a[4] = {av.x, av.y, av.z, av.w};
        float m[4], s[4];
#pragma unroll
        for (int j = 0; j < 4; ++j) {
            m[j] = (__float_as_uint(a[j]) >= T) ? 1.0f : 0.0f;
            s[j] = a[j] * m[j];
        }
        *(float4*)(acts + idx) = make_float4(s[0], s[1], s[2], s[3]);
        *(float4*)(mask + idx) = make_float4(m[0], m[1], m[2], m[3]);
    }
}

// ---------------- host side ---------------------------------------------------
extern "C" void kernel_launch(void* const* d_in, const int* in_sizes, int n_in,
                              void* d_out, int out_size, void* d_ws, size_t ws_size,
                              hipStream_t stream) {
    (void)in_sizes; (void)n_in; (void)out_size; (void)ws_size;
    const float* x = (const float*)d_in[0];
    const float* u = (const float*)d_in[1];
    float* out = (float*)d_out;                 // [0,N_TOT) sparse, [N_TOT,2N) mask
    char* ws = (char*)d_ws;
    float*    part = (float*)(ws + WS_PART);
    float*    beta = (float*)(ws + WS_BETA);
    unsigned* rank = (unsigned*)(ws + WS_RANK);
    unsigned* pref = (unsigned*)(ws + WS_PREF);
    unsigned* thr  = (unsigned*)(ws + WS_THR);
    unsigned* h1   = (unsigned*)(ws + WS_H1);
    unsigned* h2   = (unsigned*)(ws + WS_H2);
    unsigned* h3   = (unsigned*)(ws + WS_H3);

    hipMemsetAsync(d_ws, 0, WS_USED, stream);

    kSum1<<<B_DIM * SPLIT_A, 256, 0, stream>>>(x, part);
    kSum2<<<1, 32, 0, stream>>>(part, beta);

    kActsHist1<<<B_DIM * SPLIT_H, 256, 0, stream>>>(x, u, beta, out, h1);
    kScan<<<B_DIM, 256, 0, stream>>>(h1, rank, pref, thr, 4096, 1);

    kHistL<<<B_DIM * SPLIT_H, 256, 0, stream>>>(out, pref, h2, 4096, 20, 8, 0xFFFu);
    kScan<<<B_DIM, 256, 0, stream>>>(h2, rank, pref, thr, 4096, 2);

    kHistL<<<B_DIM * SPLIT_H, 256, 0, stream>>>(out, pref, h3, 256, 8, 0, 0xFFu);
    kScan<<<B_DIM, 256, 0, stream>>>(h3, rank, pref, thr, 256, 3);

    kFinal<<<B_DIM * SPLIT_H, 256, 0, stream>>>(out, thr, out + N_TOT);
}